// MultiConvAttention_52097953300623
// MI455X (gfx1250) — compile-verified
//
#include <hip/hip_runtime.h>
#include <math.h>

typedef __attribute__((ext_vector_type(2))) float v2f;
typedef __attribute__((ext_vector_type(8))) float v8f;
typedef __attribute__((ext_vector_type(2))) int   v2i;

#define LQ  32     // sequence length
#define NB  2      // batch
#define CM  128    // d_model
#define NH  4      // heads
#define DKH 32     // d_k per head
#define SW  16     // spatial width/height
#define SP  256    // 16*16 pixels

#define AS1 __attribute__((address_space(1)))
#define AS3 __attribute__((address_space(3)))

#if defined(__has_builtin)
#  if __has_builtin(__builtin_amdgcn_global_load_async_to_lds_b64)
#    define CDNA5_ASYNC_LDS 1
#  endif
#endif
#ifndef CDNA5_ASYNC_LDS
#  define CDNA5_ASYNC_LDS 0
#endif

__device__ __forceinline__ v8f wmma_f32(v2f a, v2f b, v8f c) {
    // D = A(16x4 f32) * B(4x16 f32) + C(16x16 f32)
    return __builtin_amdgcn_wmma_f32_16x16x4_f32(false, a, false, b, (short)0, c,
                                                 false, false);
}

__device__ __forceinline__ void wait_asynccnt0() {
#if defined(__has_builtin)
#  if __has_builtin(__builtin_amdgcn_s_wait_asynccnt)
    __builtin_amdgcn_s_wait_asynccnt(0);
    return;
#  endif
#endif
    asm volatile("s_wait_asynccnt 0" ::: "memory");
}

// ---------------------------------------------------------------------------
// Kernel 1: fuse conv1(1x1) with Wik/Wiv into one 128->32 matmul per head.
// Q path and both biases along the softmax axis cancel -> dropped.
// weff stored channel-pair interleaved: [(c>>1)*DKH + k]*2 + (c&1)
// ---------------------------------------------------------------------------
__global__ void k_fuse(const float* __restrict__ c1w, const float* __restrict__ c1b,
                       const float* __restrict__ wik, const float* __restrict__ wiv,
                       float* __restrict__ weffK, float* __restrict__ weffV,
                       float* __restrict__ beffK, float* __restrict__ beffV) {
    int hd  = blockIdx.x;
    int tid = threadIdx.x;
    const float* w1 = c1w + (size_t)hd * 3 * CM * CM;  // (384,128)
    const float* b1 = c1b + (size_t)hd * 3 * CM;
    const float* pk = wik + (size_t)hd * CM * DKH;     // (128,32)
    const float* pv = wiv + (size_t)hd * CM * DKH;
    for (int i = tid; i < CM * DKH; i += 256) {
        int c = i >> 5, k = i & 31;
        float sk = 0.f, sv = 0.f;
        for (int m = 0; m < CM; ++m) {
            sk += w1[(CM + m) * CM + c]     * pk[m * DKH + k];
            sv += w1[(2 * CM + m) * CM + c] * pv[m * DKH + k];
        }
        size_t o = (size_t)hd * CM * DKH + (((c >> 1) * DKH + k) << 1) + (c & 1);
        weffK[o] = sk;
        weffV[o] = sv;
    }
    if (tid < DKH) {
        int k = tid;
        float sk = 0.f, sv = 0.f;
        for (int m = 0; m < CM; ++m) {
            sk += b1[CM + m]     * pk[m * DKH + k];
            sv += b1[2 * CM + m] * pv[m * DKH + k];
        }
        beffK[hd * DKH + k] = sk;
        beffV[hd * DKH + k] = sv;
    }
}

// ---------------------------------------------------------------------------
// Kernel 1b: channel-pair transpose of x so GEMM A-operands are single b64
// loads.  xp[(l*NB+b)][(c>>1)*SP + pix] = {x[c][pix], x[c+1][pix]}
// ---------------------------------------------------------------------------
__global__ void k_xpose(const float* __restrict__ x, float* __restrict__ xp) {
    int j = blockIdx.x * 256 + threadIdx.x;   // NB*LQ * 64 * 256 = 1048576
    int pix = j & 255;
    int cp  = (j >> 8) & 63;
    int s   = j >> 14;
    const float* xs = x + (size_t)s * CM * SP;
    v2f v;
    v[0] = xs[(2 * cp) * SP + pix];
    v[1] = xs[(2 * cp + 1) * SP + pix];
    ((v2f*)xp)[(size_t)s * (CM / 2) * SP + cp * SP + pix] = v;
}

// ---------------------------------------------------------------------------
// Kernel 2: K/V projection GEMM per (head,b,l) slice via f32 WMMA.
// Single pass: each wave keeps 8 accumulators (K/V x 2 N-tiles x 2 h-rows)
// live so no operand is loop-invariant -> nothing for LICM to spill.
// K stored pair-interleaved slice-major (for conv staging),
// V stored [hd][b][k][pix][l] (l innermost, for softmax stage).
// ---------------------------------------------------------------------------
__global__ void k_proj(const float* __restrict__ xp,
                       const float* __restrict__ weffK, const float* __restrict__ weffV,
                       const float* __restrict__ beffK, const float* __restrict__ beffV,
                       float* __restrict__ Kp, float* __restrict__ Vp) {
    __shared__ v2f sWK[(CM / 2) * DKH];   // 16 KB
    __shared__ v2f sWV[(CM / 2) * DKH];   // 16 KB
    int blk = blockIdx.x;            // hd*64 + b*32 + l
    int l  = blk & 31;
    int b  = (blk >> 5) & 1;
    int hd = blk >> 6;
    int tid = threadIdx.x;
    const v2f* gK = (const v2f*)weffK + (size_t)hd * (CM / 2) * DKH;
    const v2f* gV = (const v2f*)weffV + (size_t)hd * (CM / 2) * DKH;
    for (int i = tid; i < (CM / 2) * DKH; i += 256) {
        sWK[i] = gK[i];
        sWV[i] = gV[i];
    }
    __syncthreads();
    int wave = tid >> 5, lane = tid & 31;
    int lh = lane >> 4, lm = lane & 15;
    const v2f* xs = (const v2f*)xp + (size_t)(l * NB + b) * (CM / 2) * SP;
    int h0 = wave, h1 = wave + 8;
    v8f k00 = {}, k01 = {}, v00 = {}, v01 = {};   // h0: K0,K1,V0,V1
    v8f k10 = {}, k11 = {}, v10 = {}, v11 = {};   // h1: K0,K1,V0,V1
    for (int ks = 0; ks < 32; ++ks) {
        int cp = ks * 2 + lh;                     // channel pair index
        v2f A0  = xs[cp * SP + h0 * SW + lm];
        v2f A1  = xs[cp * SP + h1 * SW + lm];
        v2f Bk0 = sWK[cp * DKH + lm];
        v2f Bk1 = sWK[cp * DKH + 16 + lm];
        v2f Bv0 = sWV[cp * DKH + lm];
        v2f Bv1 = sWV[cp * DKH + 16 + lm];
        k00 = wmma_f32(A0, Bk0, k00);
        k01 = wmma_f32(A0, Bk1, k01);
        v00 = wmma_f32(A0, Bv0, v00);
        v01 = wmma_f32(A0, Bv1, v01);
        k10 = wmma_f32(A1, Bk0, k10);
        k11 = wmma_f32(A1, Bk1, k11);
        v10 = wmma_f32(A1, Bv0, v10);
        v11 = wmma_f32(A1, Bv1, v11);
    }
    float* kslice = Kp + (size_t)((hd * NB + b) * LQ + l) * DKH * SP;
    size_t vb = ((size_t)(hd * NB + b)) * DKH;
    float bk0 = beffK[hd * DKH + lm],      bk1 = beffK[hd * DKH + 16 + lm];
    float bv0 = beffV[hd * DKH + lm],      bv1 = beffV[hd * DKH + 16 + lm];
    int k0 = lm, k1 = 16 + lm;
    for (int r = 0; r < 8; ++r) {
        int p0 = h0 * SW + r + 8 * lh;
        int p1 = h1 * SW + r + 8 * lh;
        kslice[(((k0 >> 1) * SP + p0) << 1) + (k0 & 1)] = k00[r] + bk0;
        kslice[(((k1 >> 1) * SP + p0) << 1) + (k1 & 1)] = k01[r] + bk1;
        kslice[(((k0 >> 1) * SP + p1) << 1) + (k0 & 1)] = k10[r] + bk0;
        kslice[(((k1 >> 1) * SP + p1) << 1) + (k1 & 1)] = k11[r] + bk1;
        Vp[((vb + k0) * SP + p0) * LQ + l] = v00[r] + bv0;
        Vp[((vb + k1) * SP + p0) * LQ + l] = v01[r] + bv1;
        Vp[((vb + k0) * SP + p1) * LQ + l] = v10[r] + bv0;
        Vp[((vb + k1) * SP + p1) * LQ + l] = v11[r] + bv1;
    }
}

// ---------------------------------------------------------------------------
// Kernel 3: 5x5 'SAME' conv on each K slice as 25 shifted 32x32 WMMA GEMMs.
// Pair-interleaved zero-padded slice (16 pairs x 20 x 20 x v2f) staged in LDS
// via CDNA5 async global->LDS b64 DMA (ASYNCcnt path) when available.
// ---------------------------------------------------------------------------
__global__ void k_conv(const float* __restrict__ Kp, const float* __restrict__ c2w,
                       float* __restrict__ CK) {
    __shared__ v2f sK[(DKH / 2) * 400];  // padded slice, 51.2 KB
    __shared__ v2f sW[(DKH / 2) * DKH];  // per-tap weights [ci pair][ko], 4 KB
    int blk = blockIdx.x;
    int l  = blk & 31;
    int b  = (blk >> 5) & 1;
    int hd = blk >> 6;
    int tid = threadIdx.x;
    // zero only the halo cells (async interior writes never alias them)
    for (int i = tid; i < (DKH / 2) * 400; i += 256) {
        int cell = i % 400;
        int y = cell / 20, xx = cell % 20;
        if (y < 2 || y >= 18 || xx < 2 || xx >= 18) {
            v2f z = {0.f, 0.f};
            sK[i] = z;
        }
    }
    const v2f* kin = (const v2f*)Kp + (size_t)((hd * NB + b) * LQ + l) * (DKH * SP / 2);
#if CDNA5_ASYNC_LDS
    for (int i = tid; i < (DKH / 2) * SP; i += 256) {
        int cp = i >> 8, pix = i & 255;
        int y = (pix >> 4) + 2, xx = (pix & 15) + 2;
        __builtin_amdgcn_global_load_async_to_lds_b64(
            (AS1 v2i*)(kin + cp * SP + pix),
            (AS3 v2i*)&sK[cp * 400 + y * 20 + xx], 0, 0);
    }
    wait_asynccnt0();
#else
    for (int i = tid; i < (DKH / 2) * SP; i += 256) {
        int cp = i >> 8, pix = i & 255;
        int y = (pix >> 4) + 2, xx = (pix & 15) + 2;
        sK[cp * 400 + y * 20 + xx] = kin[cp * SP + pix];
    }
#endif
    int wave = tid >> 5, lane = tid & 31;
    int lh = lane >> 4, lm = lane & 15;
    int h0 = wave, h1 = wave + 8;
    v8f a00 = {}, a01 = {}, a10 = {}, a11 = {};  // [hrow][ko-tile]
    const float* w2 = c2w + (size_t)hd * DKH * DKH * 25;  // (ko, ki, 5, 5)
    for (int tap = 0; tap < 25; ++tap) {
        int dy = tap / 5, dx = tap % 5;
        __syncthreads();  // covers slice staging on tap 0, sW reuse otherwise
        for (int i = tid; i < (DKH / 2) * DKH; i += 256) {
            int ko = i & 31, cp = i >> 5;
            v2f wv;
            wv[0] = w2[((size_t)ko * DKH + 2 * cp) * 25 + tap];
            wv[1] = w2[((size_t)ko * DKH + 2 * cp + 1) * 25 + tap];
            sW[cp * DKH + ko] = wv;
        }
        __syncthreads();
        for (int kq = 0; kq < 8; ++kq) {
            int cp = kq * 2 + lh;
            v2f A0 = sK[cp * 400 + (h0 + dy) * 20 + lm + dx];
            v2f A1 = sK[cp * 400 + (h1 + dy) * 20 + lm + dx];
            v2f B0 = sW[cp * DKH + lm];
            v2f B1 = sW[cp * DKH + 16 + lm];
            a00 = wmma_f32(A0, B0, a00);
            a01 = wmma_f32(A0, B1, a01);
            a10 = wmma_f32(A1, B0, a10);
            a11 = wmma_f32(A1, B1, a11);
        }
    }
    size_t base = ((size_t)(hd * NB + b)) * DKH;  // CK [hd][b][k][pix][l]
    for (int r = 0; r < 8; ++r) {
        int m = r + 8 * lh;
        CK[((base + lm)      * SP + h0 * SW + m) * LQ + l] = a00[r];
        CK[((base + 16 + lm) * SP + h0 * SW + m) * LQ + l] = a01[r];
        CK[((base + lm)      * SP + h1 * SW + m) * LQ + l] = a10[r];
        CK[((base + 16 + lm) * SP + h1 * SW + m) * LQ + l] = a11[r];
    }
}

// ---------------------------------------------------------------------------
// Kernel 4: softmax over l (32 contiguous values) + weighted sum with V.
// b128 vector loads; one thread per (head,b,k,pix).
// ---------------------------------------------------------------------------
__global__ void k_attn(const float* __restrict__ CK, const float* __restrict__ Vp,
                       float* __restrict__ Ocat) {
    int t = blockIdx.x * 256 + threadIdx.x;  // (hd,b,k,pix) = 65536
    int pix = t & 255;
    int k   = (t >> 8) & 31;
    int b   = (t >> 13) & 1;
    int hd  = t >> 14;
    const float4* ck4 = (const float4*)(CK + (size_t)t * LQ);
    const float4* vv4 = (const float4*)(Vp + (size_t)t * LQ);
    float4 cv[8];
    float mx = -3.4e38f;
    for (int i = 0; i < 8; ++i) {
        cv[i] = ck4[i];
        mx = fmaxf(mx, fmaxf(fmaxf(cv[i].x, cv[i].y), fmaxf(cv[i].z, cv[i].w)));
    }
    float den = 0.f, num = 0.f;
    for (int i = 0; i < 8; ++i) {
        float4 vv = vv4[i];
        float e0 = expf(cv[i].x - mx), e1 = expf(cv[i].y - mx);
        float e2 = expf(cv[i].z - mx), e3 = expf(cv[i].w - mx);
        den += (e0 + e1) + (e2 + e3);
        num += e0 * vv.x + e1 * vv.y + e2 * vv.z + e3 * vv.w;
    }
    Ocat[((size_t)(b * CM + hd * DKH + k)) * SP + pix] = num / den;
}

// ---------------------------------------------------------------------------
// Kernel 5: final W0 GEMM (512 x 128) @ (128 x 128) via f32 WMMA;
// result independent of l -> broadcast-store (two b128 per lane per slice).
// ---------------------------------------------------------------------------
__global__ void k_out(const float* __restrict__ Ocat, const float* __restrict__ W0,
                      float* __restrict__ out) {
    int wid  = blockIdx.x * 8 + (threadIdx.x >> 5);  // 256 tiles, 1 per wave
    int lane = threadIdx.x & 31;
    int lh = lane >> 4, lm = lane & 15;
    int mt = wid & 31;   // 32 M-tiles over (b, pix)
    int nt = wid >> 5;   // 8 N-tiles over d
    int b = mt >> 4;
    int pixbase = (mt & 15) * 16;
    v8f acc = {};
    for (int ks = 0; ks < 32; ++ks) {
        int c = ks * 4 + 2 * lh;
        v2f A, B;
        A[0] = Ocat[((size_t)(b * CM + c + 0)) * SP + pixbase + lm];
        A[1] = Ocat[((size_t)(b * CM + c + 1)) * SP + pixbase + lm];
        B[0] = W0[(c + 0) * CM + nt * 16 + lm];
        B[1] = W0[(c + 1) * CM + nt * 16 + lm];
        acc = wmma_f32(A, B, acc);
    }
    int d = nt * 16 + lm;
    float4 lo = make_float4(acc[0], acc[1], acc[2], acc[3]);
    float4 hi = make_float4(acc[4], acc[5], acc[6], acc[7]);
    for (int l = 0; l < LQ; ++l) {
        float* o = out + ((size_t)(l * NB + b) * CM + d) * SP + pixbase + 8 * lh;
        *(float4*)(o + 0) = lo;
        *(float4*)(o + 4) = hi;
    }
}

extern "C" void kernel_launch(void* const* d_in, const int* in_sizes, int n_in,
                              void* d_out, int out_size, void* d_ws, size_t ws_size,
                              hipStream_t stream) {
    (void)in_sizes; (void)n_in; (void)out_size; (void)ws_size;
    const float* x   = (const float*)d_in[0];
    const float* c1w = (const float*)d_in[1];
    const float* c1b = (const float*)d_in[2];
    const float* c2w = (const float*)d_in[3];
    // d_in[4] = conv2_b and d_in[5] = Wiq are mathematically eliminated
    // (constant shifts along the softmax axis cancel).
    const float* wik = (const float*)d_in[6];
    const float* wiv = (const float*)d_in[7];
    const float* w0  = (const float*)d_in[8];
    float* out = (float*)d_out;

    float* ws    = (float*)d_ws;
    float* weffK = ws;                                   // 4*128*32 (interleaved)
    float* weffV = weffK + (size_t)NH * CM * DKH;
    float* beffK = weffV + (size_t)NH * CM * DKH;        // 4*32
    float* beffV = beffK + NH * DKH;
    float* xp    = beffV + NH * DKH;                     // 64*64*256*2 (pair-interleaved x)
    float* Kp    = xp + (size_t)NB * LQ * CM * SP;       // 4*2*32*32*256 (interleaved)
    float* Vp    = Kp + (size_t)NH * NB * LQ * DKH * SP;
    float* CK    = Vp + (size_t)NH * NB * LQ * DKH * SP;
    float* Ocat  = CK + (size_t)NH * NB * LQ * DKH * SP; // 2*128*256

    k_fuse<<<NH, 256, 0, stream>>>(c1w, c1b, wik, wiv, weffK, weffV, beffK, beffV);
    k_xpose<<<(NB * LQ * (CM / 2) * SP) / 256, 256, 0, stream>>>(x, xp);
    k_proj<<<NH * NB * LQ, 256, 0, stream>>>(xp, weffK, weffV, beffK, beffV, Kp, Vp);
    k_conv<<<NH * NB * LQ, 256, 0, stream>>>(Kp, c2w, CK);
    k_attn<<<256, 256, 0, stream>>>(CK, Vp, Ocat);
    k_out<<<32, 256, 0, stream>>>(Ocat, w0, out);
}